// DiceLoss_16355235463504
// MI455X (gfx1250) — compile-verified
//
#include <hip/hip_runtime.h>
#include <stdint.h>

// ---------------- problem constants ----------------
#define NCLS   150
#define HW     (512u * 512u)          // per-class plane, elements
#define PPB    256                    // pixels per block (one per thread)
#define NPIX   (4u * HW)              // total pixels
#define NBLK   (NPIX / PPB)           // 4096 blocks
#define ROWF   257                    // LDS row stride in floats (256 + 1 TDM pad)
#define TILE_OFF 2048                 // byte offset of tile inside dynamic LDS
#define SMEM_BYTES (TILE_OFF + NCLS * ROWF * 4)   // 156,248 B < 160 KB -> 2 blocks/WGP
#define SMOOTH 1e-5f

// ws layout (floats): [0..149] xsum, [160..309] intersection, [320..469] ysum
#define WS_X 0
#define WS_I 160
#define WS_Y 320

typedef unsigned int       u32;
typedef unsigned long long u64;
typedef u32 v4u __attribute__((ext_vector_type(4)));
typedef int v8i __attribute__((ext_vector_type(8)));
typedef int v4i __attribute__((ext_vector_type(4)));

// ---------------- TDM: DMA a [150 x 256] f32 tile (row stride HW elems) into LDS ----------------
// D# per CDNA5 ISA.  Row pad: pad_interval=7 (256 DWORDs),
// pad_amount=0 (1 DWORD) -> LDS row stride = 257 floats (bank-conflict-free phase B).
__device__ __forceinline__ void tdm_load_tile_150x256(const float* gsrc, u32 lds_byte_off) {
  u64 ga = (u64)(uintptr_t)gsrc;
  v4u g0;
  g0[0] = 1u;                                            // count=1, user descriptor
  g0[1] = lds_byte_off;                                  // lds_addr (bytes)
  g0[2] = (u32)ga;                                       // global_addr[31:0]
  g0[3] = ((u32)(ga >> 32) & 0x01FFFFFFu) | (2u << 30);  // global_addr[56:32] | type=2
  v8i g1;
  g1[0] = (int)((2u << 16)      // data_size = 4 B
              | (1u << 20)      // pad_enable
              | (7u << 22));    // pad_interval = 256 DWORDs; pad_amount field = 0 -> 1 DWORD
  g1[1] = 0;                                             // barrier addr 0 | tensor_dim0[15:0]=0
  g1[2] = (int)((HW >> 16) | ((u32)NCLS << 16));         // tensor_dim0[31:16] | tensor_dim1[15:0]
  g1[3] = (int)((u32)PPB << 16);                         // tensor_dim1[31:16]=0 | tile_dim0=256
  g1[4] = NCLS;                                          // tile_dim1=150 | tile_dim2=0
  g1[5] = (int)HW;                                       // tensor_dim0_stride[31:0] = 262144
  g1[6] = 0;                                             // stride0[47:32] | stride1[15:0]
  g1[7] = 0;                                             // stride1[47:16]
  v4i gz  = {0, 0, 0, 0};                                // groups 2/3 unused (2-D tile)
  v8i gz8 = {0, 0, 0, 0, 0, 0, 0, 0};                    // extra group (6-arg form), zero-filled
  __builtin_amdgcn_tensor_load_to_lds(g0, g1, gz, gz, gz8, 0);
}

// ---------------- main kernel (placed first so the disasm snippet shows the TDM path) ----------------
__global__ __launch_bounds__(PPB) void dice_main(const float* __restrict__ x,
                                                 const long long* __restrict__ tgt,
                                                 float* __restrict__ ws) {
  extern __shared__ float smem[];
  float* acc_x = smem + WS_X;
  float* acc_i = smem + WS_I;
  float* acc_y = smem + WS_Y;
  float* tile  = (float*)((char*)smem + TILE_OFF);

  const int tid = threadIdx.x;
  const u32 P  = (u32)blockIdx.x * PPB;   // first pixel of this block
  const u32 b  = P >> 18;                 // batch index (HW = 1<<18)
  const u32 ip = P & (HW - 1u);           // offset inside plane

  // wave 0 issues the tensor DMA (TDM ignores EXEC -> must be wave-unique)
  if (tid < 32) {
    const float* src = x + ((size_t)b * NCLS * HW + ip);
    tdm_load_tile_150x256(src, __builtin_amdgcn_groupstaticsize() + TILE_OFF);
  }
  // all waves zero the per-block class accumulators meanwhile
  if (tid < 3 * 160) smem[tid] = 0.0f;
  if (tid < 32) __builtin_amdgcn_s_wait_tensorcnt(0);
  __syncthreads();   // tile + accumulators visible to the whole workgroup

  // ---- phase A: one pixel per lane; softmax over 150 classes from LDS ----
  long long tv = tgt[P + tid];
  int ti = (tv == 255) ? 0 : (int)tv;     // ignore_index -> class 0 (matches reference)

  float m = -3.402823466e38f;
  for (int c = 0; c < NCLS; ++c) m = fmaxf(m, tile[c * ROWF + tid]);
  float s = 0.0f;
  for (int c = 0; c < NCLS; ++c) s += __expf(tile[c * ROWF + tid] - m);
  float inv = 1.0f / s;
  float pt = 0.0f;
  for (int c = 0; c < NCLS; ++c) {
    float pr = __expf(tile[c * ROWF + tid] - m) * inv;
    if (c == ti) pt = pr;
    tile[c * ROWF + tid] = pr * pr;       // overwrite logits with p^2 for phase B
  }
  unsafeAtomicAdd(&acc_i[ti], pt);        // ds_add_f32
  unsafeAtomicAdd(&acc_y[ti], 1.0f);
  __syncthreads();

  // ---- phase B: class per lane; sum p^2 over this wave's 32 pixels (stride-257 -> no conflicts) ----
  const int w = tid >> 5, l = tid & 31;
  for (int c = l; c < NCLS; c += 32) {
    const float* row = &tile[c * ROWF + (w << 5)];
    float s2 = 0.0f;
#pragma unroll
    for (int i = 0; i < 32; ++i) s2 += row[i];
    unsafeAtomicAdd(&acc_x[c], s2);
  }
  __syncthreads();

  // ---- flush block partials: one global f32 atomic per class ----
  if (tid < NCLS) {
    unsafeAtomicAdd(&ws[WS_X + tid], acc_x[tid]);
    unsafeAtomicAdd(&ws[WS_I + tid], acc_i[tid]);
    unsafeAtomicAdd(&ws[WS_Y + tid], acc_y[tid]);
  }
}

// ---------------- helper kernels ----------------
__global__ void dice_zero(float* ws) { ws[threadIdx.x] = 0.0f; }

__global__ void dice_finalize(const float* __restrict__ ws, float* __restrict__ out) {
  __shared__ float red[256];
  const int c = threadIdx.x;
  float contrib = 0.0f;
  if (c < NCLS) {
    float d = (2.0f * ws[WS_I + c] + SMOOTH) / (ws[WS_X + c] + ws[WS_Y + c] + SMOOTH);
    out[1 + c] = d;                       // class-wise dice
    contrib = 1.0f - d;
  }
  red[c] = contrib;
  __syncthreads();
  for (int s = 128; s > 0; s >>= 1) {
    if (c < s) red[c] += red[c + s];
    __syncthreads();
  }
  if (c == 0) out[0] = red[0] / (float)NCLS;   // loss = mean(1 - dice)
}

// ---------------- launch ----------------
extern "C" void kernel_launch(void* const* d_in, const int* in_sizes, int n_in,
                              void* d_out, int out_size, void* d_ws, size_t ws_size,
                              hipStream_t stream) {
  (void)in_sizes; (void)n_in; (void)out_size; (void)ws_size;
  const float*     x   = (const float*)d_in[0];
  const long long* tgt = (const long long*)d_in[1];
  float* out = (float*)d_out;
  float* ws  = (float*)d_ws;

  (void)hipFuncSetAttribute(reinterpret_cast<const void*>(dice_main),
                            hipFuncAttributeMaxDynamicSharedMemorySize, SMEM_BYTES);

  dice_zero<<<1, 512, 0, stream>>>(ws);
  dice_main<<<NBLK, PPB, SMEM_BYTES, stream>>>(x, tgt, ws);
  dice_finalize<<<1, 256, 0, stream>>>(ws, out);
}